// LSTM_model_26302379720943
// MI455X (gfx1250) — compile-verified
//
#include <hip/hip_runtime.h>

#define EMB    512
#define HID    1024
#define BATCH  64
#define TLEN   512
#define KTOT   (EMB + HID)      // 1536
#define CHUNK  64
#define NCHUNK (KTOT / CHUNK)   // 24

typedef float v2f __attribute__((ext_vector_type(2)));
typedef float v8f __attribute__((ext_vector_type(8)));

__device__ __forceinline__ float fast_rcp(float x) { return __builtin_amdgcn_rcpf(x); }
__device__ __forceinline__ float fast_sigmoid(float x) {
    return fast_rcp(1.0f + __expf(-x));
}
__device__ __forceinline__ float fast_tanh(float x) {
    const float e = __expf(2.0f * x);
    return (e - 1.0f) * fast_rcp(e + 1.0f);
}

// Async copy of 16 bytes global -> LDS (CDNA5 GLOBAL_LOAD_ASYNC_TO_LDS_B128,
// GVS addressing: SGPR64 base + VGPR32 byte offset). Tracked by ASYNCcnt.
// ldsoff is the wave-relative LDS byte address (low 32 bits of a generic
// pointer into __shared__ memory -- the ptrtoint at the call site also forces
// the shared object to escape so the "memory" clobber really covers it).
__device__ __forceinline__ void async_copy16(unsigned ldsoff, unsigned goff,
                                             const void* base)
{
    asm volatile("global_load_async_to_lds_b128 %0, %1, %2"
                 :: "v"(ldsoff), "v"(goff), "s"(base)
                 : "memory");
}

// One timestep. Grid: 64 blocks (one per 16-col hidden tile) x 128 threads
// (4 waves = 4 batch m-tiles). All waves share identical weight fragments,
// staged in LDS via double-buffered async copies; WMMA f32 16x16x4 over
// K = EMB + HID with B fragments software-pipelined one k-step ahead.
__global__ void __launch_bounds__(128)
lstm_step_kernel(const float* __restrict__ x_emb,
                 const float* __restrict__ W_F, const float* __restrict__ b_F,
                 const float* __restrict__ W_I, const float* __restrict__ b_I,
                 const float* __restrict__ W_C, const float* __restrict__ b_C,
                 const float* __restrict__ W_O, const float* __restrict__ b_O,
                 const float* __restrict__ c_in, float* __restrict__ c_out,
                 float* __restrict__ all_h, float* __restrict__ c_fin,
                 float* __restrict__ h_fin,
                 int t, int is_last)
{
    // Double-buffered weight stage: [buf][gate][k in chunk][n] = 32 KB
    __shared__ float sB[2][4][CHUNK][16];

    const int tid   = threadIdx.x;
    const int lane  = tid & 31;
    const int wid   = tid >> 5;           // wave id == m_tile (0..3)
    const int n0    = blockIdx.x * 16;    // hidden-col tile (64 blocks)
    const int m0    = wid * 16;
    const int lhalf = lane >> 4;          // 0: lanes 0-15, 1: lanes 16-31
    const int l15   = lane & 15;

    // ---- async stage addressing (only kbase varies per chunk) ----
    const int      sk   = tid >> 2;                    // k within half-chunk (0..31)
    const int      sn   = (tid & 3) * 4;               // n within tile (0,4,8,12)
    const unsigned gcol = (unsigned)(n0 + sn) * 4u;    // global col byte offset
    const unsigned gstr = (unsigned)(CHUNK * 16 * 4);  // LDS bytes per gate slab
    const unsigned hrow = (unsigned)(32 * HID * 4);    // +32 rows, global bytes
    const unsigned hlds = (unsigned)(32 * 16 * 4);     // +32 rows, LDS bytes

    auto issue = [&](int buf, int kbase) {
        const unsigned grow  = (unsigned)((kbase + sk) * HID) * 4u + gcol;
        // real LDS address of this thread's staging slot (forces sB to escape)
        const unsigned lbase = (unsigned)(uintptr_t)&sB[buf][0][sk][sn];
        async_copy16(lbase + 0u * gstr,        grow,        W_F);
        async_copy16(lbase + 0u * gstr + hlds, grow + hrow, W_F);
        async_copy16(lbase + 1u * gstr,        grow,        W_I);
        async_copy16(lbase + 1u * gstr + hlds, grow + hrow, W_I);
        async_copy16(lbase + 2u * gstr,        grow,        W_C);
        async_copy16(lbase + 2u * gstr + hlds, grow + hrow, W_C);
        async_copy16(lbase + 3u * gstr,        grow,        W_O);
        async_copy16(lbase + 3u * gstr + hlds, grow + hrow, W_O);
    };

    auto ldfrag = [&](int buf, int g, int kk) -> v2f {
        v2f b;
        b.x = sB[buf][g][kk + lhalf * 2][l15];
        b.y = sB[buf][g][kk + lhalf * 2 + 1][l15];
        return b;
    };

    // ---- A-fragment source row (batch element) for this lane ----
    const int arow = m0 + l15;
    const float* __restrict__ xrow = x_emb + ((size_t)arow * TLEN + t) * EMB;
    const float* __restrict__ crow = c_in  + (size_t)arow * HID;

    v8f accF = {}, accI = {}, accC = {}, accO = {};

    issue(0, 0);
    for (int chunk = 0; chunk < NCHUNK; ++chunk) {
        const int buf   = chunk & 1;
        const int kbase = chunk * CHUNK;

        if (chunk + 1 < NCHUNK) {
            issue(buf ^ 1, kbase + CHUNK);
            asm volatile("s_wait_asynccnt 8" ::: "memory");  // this chunk landed
        } else {
            asm volatile("s_wait_asynccnt 0" ::: "memory");
        }
        __syncthreads();   // all waves' stage of this chunk visible

        // chunk never straddles the EMB/HID boundary (512 % 64 == 0)
        const float* __restrict__ ar =
            (kbase < EMB) ? (xrow + kbase) : (crow + (kbase - EMB));

        // software pipeline: B fragments loaded one k-step ahead
        v2f bF0 = ldfrag(buf, 0, 0);
        v2f bI0 = ldfrag(buf, 1, 0);
        v2f bC0 = ldfrag(buf, 2, 0);
        v2f bO0 = ldfrag(buf, 3, 0);

#pragma unroll
        for (int kk = 0; kk < CHUNK; kk += 4) {
            v2f bF1 = bF0, bI1 = bI0, bC1 = bC0, bO1 = bO0;
            if (kk + 4 < CHUNK) {
                bF1 = ldfrag(buf, 0, kk + 4);
                bI1 = ldfrag(buf, 1, kk + 4);
                bC1 = ldfrag(buf, 2, kk + 4);
                bO1 = ldfrag(buf, 3, kk + 4);
            }

            const int kr = kk + lhalf * 2;
            v2f a;
            a.x = ar[kr];
            a.y = ar[kr + 1];

            accF = __builtin_amdgcn_wmma_f32_16x16x4_f32(false, a, false, bF0,
                                                         (short)0, accF, false, false);
            accI = __builtin_amdgcn_wmma_f32_16x16x4_f32(false, a, false, bI0,
                                                         (short)0, accI, false, false);
            accC = __builtin_amdgcn_wmma_f32_16x16x4_f32(false, a, false, bC0,
                                                         (short)0, accC, false, false);
            accO = __builtin_amdgcn_wmma_f32_16x16x4_f32(false, a, false, bO0,
                                                         (short)0, accO, false, false);

            bF0 = bF1; bI0 = bI1; bC0 = bC1; bO0 = bO1;
        }
        __syncthreads();   // all waves done reading buf before it is refilled
    }

    // ---- fused gate math + cell update ----
    const int   n     = n0 + l15;
    const float biasF = b_F[n];
    const float biasI = b_I[n];
    const float biasC = b_C[n];
    const float biasO = b_O[n];

#pragma unroll
    for (int j = 0; j < 8; ++j) {
        const int row = m0 + j + lhalf * 8;   // C/D layout: M = j (+8 upper half)
        const float f  = fast_sigmoid(accF[j] + biasF);
        const float i  = fast_sigmoid(accI[j] + biasI);
        const float gc = fast_tanh   (accC[j] + biasC);
        const float o  = fast_sigmoid(accO[j] + biasO);
        const float cp = c_in[(size_t)row * HID + n];
        const float cn = f * cp + i * gc;
        const float h  = fast_tanh(cn) * o;

        c_out[(size_t)row * HID + n] = cn;
        all_h[((size_t)row * TLEN + t) * HID + n] = h;
        if (is_last) {
            c_fin[(size_t)row * HID + n] = cn;
            h_fin[(size_t)row * HID + n] = h;
        }
    }
}

__global__ void zero_kernel(float* __restrict__ p, int nelem)
{
    const int i = blockIdx.x * blockDim.x + threadIdx.x;
    if (i < nelem) p[i] = 0.0f;
}

extern "C" void kernel_launch(void* const* d_in, const int* in_sizes, int n_in,
                              void* d_out, int out_size, void* d_ws, size_t ws_size,
                              hipStream_t stream)
{
    (void)in_sizes; (void)n_in; (void)out_size; (void)ws_size;

    const float* x_emb = (const float*)d_in[0];
    const float* W_F   = (const float*)d_in[1];
    const float* b_F   = (const float*)d_in[2];
    const float* W_I   = (const float*)d_in[3];
    const float* b_I   = (const float*)d_in[4];
    const float* W_C   = (const float*)d_in[5];
    const float* b_C   = (const float*)d_in[6];
    const float* W_O   = (const float*)d_in[7];
    const float* b_O   = (const float*)d_in[8];

    float* all_h = (float*)d_out;                              // [B, T, H]
    float* c_fin = all_h + (size_t)BATCH * TLEN * HID;         // [B, H]
    float* h_fin = c_fin + (size_t)BATCH * HID;                // [B, H]

    // double-buffered cell state in workspace
    float* cbuf0 = (float*)d_ws;
    float* cbuf1 = cbuf0 + (size_t)BATCH * HID;

    const int celems = BATCH * HID;
    zero_kernel<<<(celems + 255) / 256, 256, 0, stream>>>(cbuf0, celems);

    for (int t = 0; t < TLEN; ++t) {
        const float* cin  = (t & 1) ? cbuf1 : cbuf0;
        float*       cout = (t & 1) ? cbuf0 : cbuf1;
        lstm_step_kernel<<<64, 128, 0, stream>>>(
            x_emb, W_F, b_F, W_I, b_I, W_C, b_C, W_O, b_O,
            cin, cout, all_h, c_fin, h_fin, t, (t == TLEN - 1) ? 1 : 0);
    }
}